// Loss_23021024706808
// MI455X (gfx1250) — compile-verified
//
#include <hip/hip_runtime.h>
#include <stdint.h>

// Problem constants (fixed by setup_inputs in the reference)
#define BB 32
#define HH 256
#define WW 512

typedef unsigned int u32;
typedef uint64_t u64;
typedef u32 v4u __attribute__((ext_vector_type(4)));
typedef int  v8i __attribute__((ext_vector_type(8)));
typedef int  v4i __attribute__((ext_vector_type(4)));

#if __has_builtin(__builtin_amdgcn_tensor_load_to_lds)
#define HAS_TDM 1
#else
#define HAS_TDM 0
#endif

typedef __attribute__((address_space(3))) float lds_float;

// ---------------------------------------------------------------- utilities

// wave32 shuffle reduction (warpSize == 32 on gfx1250)
__device__ __forceinline__ float waveReduceSum(float v) {
    for (int o = 16; o > 0; o >>= 1) v += __shfl_down(v, o, 32);
    return v;
}

// Horizontal bilinear sample with zeros padding (grid_sample semantics: each
// corner zeroed individually when out of range; y is always exact here).
__device__ __forceinline__ float sample_row(const float* __restrict__ row, int w, float xs) {
    float xf = floorf(xs);
    int   x0 = (int)xf;
    float f  = xs - xf;
    float v0 = (x0 >= 0 && x0 < w)         ? row[x0]     : 0.f;
    float v1 = (x0 + 1 >= 0 && x0 + 1 < w) ? row[x0 + 1] : 0.f;
    return v0 * (1.f - f) + v1 * f;
}

#if HAS_TDM
// Issue one TDM 2D-tile load: TW x TW f32 tile starting at gsrc (row pitch w
// elements). tensor_dim is the remaining image extent so OOB reads zero-fill.
__device__ __forceinline__ void tdm_tile_load(const float* gsrc, float* lds_dst,
                                              int rem_x, int rem_y, int w, int tw) {
    u64 gaddr = (u64)(uintptr_t)gsrc;
    u32 ldsa  = (u32)(uintptr_t)(lds_float*)lds_dst;
    u32 td0   = (u32)rem_x;
    u32 td1   = (u32)rem_y;
    v4u g0;
    g0[0] = 1u;                                              // count=1 (valid D#)
    g0[1] = ldsa;                                            // lds_addr (bytes)
    g0[2] = (u32)gaddr;                                      // global_addr[31:0]
    g0[3] = ((u32)(gaddr >> 32) & 0x01FFFFFFu) | (2u << 30); // addr[56:32] | type=2
    v8i g1;
    g1[0] = (int)(2u << 16);                                 // data_size = 4B
    g1[1] = (int)((td0 & 0xFFFFu) << 16);                    // tensor_dim0 lo16
    g1[2] = (int)(((td0 >> 16) & 0xFFFFu) | ((td1 & 0xFFFFu) << 16));
    g1[3] = (int)(((td1 >> 16) & 0xFFFFu) | ((u32)tw << 16)); // tile_dim0
    g1[4] = (int)(u32)tw;                                     // tile_dim1
    g1[5] = (int)(u32)w;                                      // tensor_dim0_stride lo32
    g1[6] = 0;
    g1[7] = 0;
    v4i g2 = {0, 0, 0, 0};
    v4i g3 = {0, 0, 0, 0};
    v8i g4 = {0, 0, 0, 0, 0, 0, 0, 0};
    __builtin_amdgcn_tensor_load_to_lds(g0, g1, g2, g3, g4, 0);
}
#endif

// ---------------------------------------------------------------- kernels

__global__ void k_zero(float* acc) {
    int i = threadIdx.x;
    if (i < 32) acc[i] = 0.f;
}

// align_corners=True bilinear resize, one thread per output element
__global__ void k_resize(const float* __restrict__ src, float* __restrict__ dst,
                         int Hs, int Ws, int Hd, int Wd) {
    int idx = blockIdx.x * blockDim.x + threadIdx.x;
    int total = BB * 3 * Hd * Wd;
    if (idx >= total) return;
    int x = idx % Wd;
    int t = idx / Wd;
    int y = t % Hd;
    int bc = t / Hd;
    float sy = (float)(Hs - 1) / (float)(Hd - 1);
    float sx = (float)(Ws - 1) / (float)(Wd - 1);
    float fy = y * sy, fx = x * sx;
    int y0 = (int)floorf(fy); int y1 = min(y0 + 1, Hs - 1); float wy = fy - (float)y0;
    int x0 = (int)floorf(fx); int x1 = min(x0 + 1, Ws - 1); float wx = fx - (float)x0;
    const float* p = src + (size_t)bc * Hs * Ws;
    float top = p[(size_t)y0 * Ws + x0] * (1.f - wx) + p[(size_t)y0 * Ws + x1] * wx;
    float bot = p[(size_t)y1 * Ws + x0] * (1.f - wx) + p[(size_t)y1 * Ws + x1] * wx;
    dst[idx] = top * (1.f - wy) + bot * wy;
}

// Fused per-level pointwise pass: both warps computed on the fly.
// acc[0] sum|lp-left_est|, acc[1] sum|rp-right_est|,
// acc[2] sum|dl-r2l|,      acc[3] sum|dr-l2r|,
// acc[4] smooth_l sum,     acc[5] smooth_r sum
__global__ void k_warp_pw(const float* __restrict__ lp, const float* __restrict__ rp,
                          const float* __restrict__ disp, float* __restrict__ acc,
                          int h, int w) {
    __shared__ float spart[8 * 6];   // 8 waves x 6 accumulators
    int idx = blockIdx.x * blockDim.x + threadIdx.x;
    int plane = h * w;
    int total = BB * plane;
    float vals[6] = {0.f, 0.f, 0.f, 0.f, 0.f, 0.f};
    if (idx < total) {
        int x = idx % w;
        int t = idx / w;
        int y = t % h;
        int b = t / h;
        const float* dlp = disp + (size_t)b * 2 * plane;
        const float* drp = dlp + plane;
        float dl = dlp[(size_t)y * w + x];
        float dr = drp[(size_t)y * w + x];
        float xsl = (float)x - dl * (float)(w - 1);  // left_est samples right img
        float xsr = (float)x + dr * (float)(w - 1);  // right_est samples left img
        float gxa_l = 0.f, gya_l = 0.f, gxa_r = 0.f, gya_r = 0.f;
        for (int c = 0; c < 3; ++c) {
            const float* lrow = lp + ((size_t)(b * 3 + c) * h + y) * w;
            const float* rrow = rp + ((size_t)(b * 3 + c) * h + y) * w;
            __builtin_prefetch(lrow + x, 0, 0);   // global_prefetch_b8
            float le = sample_row(rrow, w, xsl);
            float re = sample_row(lrow, w, xsr);
            vals[0] += fabsf(lrow[x] - le);
            vals[1] += fabsf(rrow[x] - re);
            if (x < w - 1) {
                gxa_l += fabsf(lrow[x] - lrow[x + 1]);
                gxa_r += fabsf(rrow[x] - rrow[x + 1]);
            }
            if (y < h - 1) {
                gya_l += fabsf(lrow[x] - lrow[x + w]);
                gya_r += fabsf(rrow[x] - rrow[x + w]);
            }
        }
        vals[2] = fabsf(dl - sample_row(drp + (size_t)y * w, w, xsl));
        vals[3] = fabsf(dr - sample_row(dlp + (size_t)y * w, w, xsr));
        float gxd_l = (x < w - 1) ? dl - dlp[(size_t)y * w + x + 1]   : 0.f;
        float gyd_l = (y < h - 1) ? dl - dlp[(size_t)(y + 1) * w + x] : 0.f;
        float gxd_r = (x < w - 1) ? dr - drp[(size_t)y * w + x + 1]   : 0.f;
        float gyd_r = (y < h - 1) ? dr - drp[(size_t)(y + 1) * w + x] : 0.f;
        const float third = 1.f / 3.f;
        vals[4] = fabsf(gxd_l) * expf(-gxa_l * third) + fabsf(gyd_l) * expf(-gya_l * third);
        vals[5] = fabsf(gxd_r) * expf(-gxa_r * third) + fabsf(gyd_r) * expf(-gya_r * third);
    }
    int lane = threadIdx.x & 31;
    int wv   = threadIdx.x >> 5;
    for (int k = 0; k < 6; ++k) {
        float r = waveReduceSum(vals[k]);
        if (lane == 0) spart[wv * 6 + k] = r;
    }
    __syncthreads();
    if (threadIdx.x < 6) {
        float s = 0.f;
        for (int wvi = 0; wvi < 8; ++wvi) s += spart[wvi * 6 + threadIdx.x];
        atomicAdd(&acc[threadIdx.x], s);
    }
}

// DSSIM over (base, warped(wsrc)) for one level/side.
// 16x16 outputs per block; 18x18 input windows staged in LDS.
// Both the 'x' tile and the disparity tile are fetched by the Tensor Data
// Mover; only the data-dependent warped gather stays on the VALU/VMEM path.
#define TS 16
#define TW 18
__global__ void k_dssim(const float* __restrict__ base, const float* __restrict__ wsrc,
                        const float* __restrict__ disp, int chan_off, float sgn,
                        int h, int w, float* __restrict__ accp) {
    __shared__ float sX[TW * TW];
    __shared__ float sD[TW * TW];
    __shared__ float sY[TW * TW];
    __shared__ float spart[8];
    int tid = threadIdx.x;
    int bz = blockIdx.z;
    int b = bz / 3, c = bz % 3;
    int tx0 = blockIdx.x * TS, ty0 = blockIdx.y * TS;
    int plane = h * w;
    const float* bimg = base + ((size_t)(b * 3 + c) * h) * w;
    const float* wimg = wsrc + ((size_t)(b * 3 + c) * h) * w;
    const float* dch  = disp + (size_t)b * 2 * plane + chan_off;

#if HAS_TDM
    if (tid < 32) {   // wave 0 issues both TDM descriptors
        tdm_tile_load(bimg + (size_t)ty0 * w + tx0, sX, w - tx0, h - ty0, w, TW);
        tdm_tile_load(dch  + (size_t)ty0 * w + tx0, sD, w - tx0, h - ty0, w, TW);
    }
#if __has_builtin(__builtin_amdgcn_s_wait_tensorcnt)
    __builtin_amdgcn_s_wait_tensorcnt(0);
#else
    asm volatile("s_wait_tensorcnt 0x0" ::: "memory");
#endif
    __syncthreads();
#else
    for (int i = tid; i < TW * TW; i += 256) {
        int iy = i / TW, ix = i % TW;
        int yy = min(ty0 + iy, h - 1), xx = min(tx0 + ix, w - 1);
        sX[i] = bimg[(size_t)yy * w + xx];
        sD[i] = dch[(size_t)yy * w + xx];
    }
    __syncthreads();
#endif

    // Warped estimate tile (data-dependent gather; disp comes from LDS).
    // OOB tile slots hold zero-filled d and are never consumed by valid
    // SSIM windows; sample_row bounds-checks keep all reads in range.
    for (int i = tid; i < TW * TW; i += 256) {
        int iy = i / TW, ix = i % TW;
        int yy = min(ty0 + iy, h - 1);
        int xx = tx0 + ix;
        float d  = sD[i];
        float xs = (float)xx + sgn * d * (float)(w - 1);
        sY[i] = sample_row(wimg + (size_t)yy * w, w, xs);
    }
    __syncthreads();

    int lx = tid % TS, ly = tid / TS;
    int ox = tx0 + lx, oy = ty0 + ly;
    float val = 0.f;
    if (ox <= w - 3 && oy <= h - 3) {
        float sx_ = 0.f, sy_ = 0.f, sxx = 0.f, syy = 0.f, sxy = 0.f;
        for (int dy = 0; dy < 3; ++dy)
            for (int dx = 0; dx < 3; ++dx) {
                float a  = sX[(ly + dy) * TW + (lx + dx)];
                float bb = sY[(ly + dy) * TW + (lx + dx)];
                sx_ += a; sy_ += bb;
                sxx += a * a; syy += bb * bb; sxy += a * bb;
            }
        const float inv9 = 1.f / 9.f;
        float mx = sx_ * inv9, my = sy_ * inv9;
        float vx = sxx * inv9 - mx * mx;
        float vy = syy * inv9 - my * my;
        float cv = sxy * inv9 - mx * my;
        const float C1c = 0.0001f, C2c = 0.0009f;
        float ssim = ((2.f * mx * my + C1c) * (2.f * cv + C2c)) /
                     ((mx * mx + my * my + C1c) * (vx + vy + C2c));
        val = fminf(fmaxf((1.f - ssim) * 0.5f, 0.f), 1.f);
    }
    float r = waveReduceSum(val);
    int lane = tid & 31, wv = tid >> 5;
    if (lane == 0) spart[wv] = r;
    __syncthreads();
    if (tid == 0) {
        float s = 0.f;
        for (int wvi = 0; wvi < 8; ++wvi) s += spart[wvi];
        atomicAdd(accp, s);
    }
}

// Combine per-level accumulators into (total, AP, LR, DS)
__global__ void k_final(const float* __restrict__ acc, float* __restrict__ out) {
    if (threadIdx.x != 0 || blockIdx.x != 0) return;
    float AP = 0.f, LR = 0.f, DS = 0.f;
    for (int l = 0; l < 4; ++l) {
        int h = HH >> l, w = WW >> l;
        float ci = (float)((u64)BB * 3 * h * w);
        float cd = (float)((u64)BB * h * w);
        float cs = (float)((u64)BB * 3 * (h - 2) * (w - 2));
        const float* a = acc + l * 8;
        AP += 0.85f * (a[6] / cs) + 0.15f * (a[0] / ci);
        AP += 0.85f * (a[7] / cs) + 0.15f * (a[1] / ci);
        LR += a[2] / cd + a[3] / cd;
        DS += (a[4] / cd + a[5] / cd) / (float)(1 << l);
    }
    AP *= 0.85f;   // outer *alpha_AP as in reference
    DS *= 0.1f;
    out[0] = AP + LR + DS;
    out[1] = AP;
    out[2] = LR;
    out[3] = DS;
}

// ---------------------------------------------------------------- launch

extern "C" void kernel_launch(void* const* d_in, const int* in_sizes, int n_in,
                              void* d_out, int out_size, void* d_ws, size_t ws_size,
                              hipStream_t stream) {
    (void)in_sizes; (void)n_in; (void)out_size; (void)ws_size;
    const float* disp[4] = {(const float*)d_in[0], (const float*)d_in[1],
                            (const float*)d_in[2], (const float*)d_in[3]};
    const float* left  = (const float*)d_in[4];
    const float* right = (const float*)d_in[5];

    float* ws  = (float*)d_ws;
    float* acc = ws;  // 32 accumulators: [level*8 + term]

    // Pyramid storage in workspace (levels 1..3); level 0 aliases the inputs.
    size_t S[4] = {(size_t)BB * 3 * HH * WW,
                   (size_t)BB * 3 * (HH / 2) * (WW / 2),
                   (size_t)BB * 3 * (HH / 4) * (WW / 4),
                   (size_t)BB * 3 * (HH / 8) * (WW / 8)};
    const float* lp[4];
    const float* rp[4];
    lp[0] = left; rp[0] = right;
    size_t off = 64;  // skip accumulator block (padded)
    float* lpm[4], *rpm[4];
    for (int l = 1; l < 4; ++l) {
        lpm[l] = ws + off; off += S[l];
        rpm[l] = ws + off; off += S[l];
        lp[l] = lpm[l]; rp[l] = rpm[l];
    }

    k_zero<<<1, 64, 0, stream>>>(acc);

    for (int l = 1; l < 4; ++l) {
        int Hs = HH >> (l - 1), Ws = WW >> (l - 1), Hd = HH >> l, Wd = WW >> l;
        int tot = BB * 3 * Hd * Wd;
        int blk = (tot + 255) / 256;
        k_resize<<<blk, 256, 0, stream>>>(lp[l - 1], lpm[l], Hs, Ws, Hd, Wd);
        k_resize<<<blk, 256, 0, stream>>>(rp[l - 1], rpm[l], Hs, Ws, Hd, Wd);
    }

    for (int l = 0; l < 4; ++l) {
        int h = HH >> l, w = WW >> l;
        int tot = BB * h * w;
        k_warp_pw<<<(tot + 255) / 256, 256, 0, stream>>>(lp[l], rp[l], disp[l],
                                                         acc + l * 8, h, w);
        int gx = (w - 2 + TS - 1) / TS, gy = (h - 2 + TS - 1) / TS;
        dim3 g((unsigned)gx, (unsigned)gy, (unsigned)(BB * 3));
        // left side: x=lp, est = warp(rp, -dl)
        k_dssim<<<g, 256, 0, stream>>>(lp[l], rp[l], disp[l], 0, -1.f, h, w,
                                       acc + l * 8 + 6);
        // right side: x=rp, est = warp(lp, +dr)
        k_dssim<<<g, 256, 0, stream>>>(rp[l], lp[l], disp[l], h * w, +1.f, h, w,
                                       acc + l * 8 + 7);
    }

    k_final<<<1, 1, 0, stream>>>(acc, (float*)d_out);
}